// EnhancedReflectiveCognitiveGraph_47665547051325
// MI455X (gfx1250) — compile-verified
//
#include <hip/hip_runtime.h>
#include <hip/hip_bf16.h>

#define NNODES 50000
#define NEDGES 600000
#define NFEAT  128
#define NHEAD  8

typedef float v2f __attribute__((ext_vector_type(2)));
typedef float v8f __attribute__((ext_vector_type(8)));
typedef unsigned int u32x4 __attribute__((ext_vector_type(4)));
typedef int i32x4 __attribute__((ext_vector_type(4)));
typedef int i32x8 __attribute__((ext_vector_type(8)));

// monotonic float <-> uint mapping so GLOBAL_ATOMIC_MAX_U32 implements float max
__device__ __forceinline__ unsigned f2ord(float f) {
  unsigned u = __float_as_uint(f);
  return (u & 0x80000000u) ? ~u : (u | 0x80000000u);
}
__device__ __forceinline__ float ord2f(unsigned u) {
  return (u & 0x80000000u) ? __uint_as_float(u & 0x7FFFFFFFu) : __uint_as_float(~u);
}

// Issue a TDM DMA of a 16x128 f32 row-tile (8KB) from `gbase` into LDS offset 0.
// D# per CDNA5 ISA: 2-D tensor, data_size=4B,
// tensor_dim0=128 (contig), tensor_dim1=16 rows, tile==tensor, stride0=128.
__device__ __forceinline__ void tdm_load_tile16x128(const float* gbase) {
  unsigned long long ga = (unsigned long long)gbase;
  u32x4 g0;
  g0[0] = 1u;                                   // count=1, user descriptor
  g0[1] = 0u;                                   // lds_addr = 0 (sole shared buf)
  g0[2] = (unsigned)ga;                         // global_addr[31:0]
  g0[3] = (unsigned)((ga >> 32) & 0x01FFFFFFu)  // global_addr[56:32]
          | 0x80000000u;                        // type=2 ("image")
  i32x8 g1;
  g1[0] = 0x00020000;                           // data_size=2 (4 bytes)
  g1[1] = (int)(128u << 16);                    // tensor_dim0 = 128 (lo16 @ bit48)
  g1[2] = (int)(16u << 16);                     // tensor_dim1 = 16  (lo16 @ bit80)
  g1[3] = (int)(128u << 16);                    // tile_dim0 = 128   (@ bit112)
  g1[4] = 16;                                   // tile_dim1 = 16    (@ bit128)
  g1[5] = 128;                                  // tensor_dim0_stride = 128
  g1[6] = 0;
  g1[7] = 0;
  i32x4 gz4 = {0, 0, 0, 0};
  i32x8 gz8 = {0, 0, 0, 0, 0, 0, 0, 0};
  // 6-arg form (upstream clang-23 / therock-10.0 HIP headers)
  __builtin_amdgcn_tensor_load_to_lds(g0, g1, gz4, gz4, gz8, 0);
  __builtin_amdgcn_s_wait_tensorcnt(0);
}

// Opaque never-taken store: stops GlobalOpt treating the (TDM-written) LDS
// buffer as store-free and folding ds_loads to undef. Zero runtime cost.
#define LDS_KEEPALIVE(buf)                                    \
  do {                                                        \
    unsigned _zz;                                             \
    asm volatile("s_mov_b32 %0, 0" : "=s"(_zz));              \
    if (_zz) (buf)[threadIdx.x & 2047] = 0.0f;                \
  } while (0)

// ---------------------------------------------------------------- init
__global__ void gat_init(float* __restrict__ agg, unsigned* __restrict__ segmax,
                         float* __restrict__ segsum) {
  int t = blockIdx.x * blockDim.x + threadIdx.x;
  if (t < NNODES * NFEAT) agg[t] = 0.0f;
  if (t < NNODES * NHEAD) {
    segmax[t] = 0x007FFFFFu;  // f2ord(-inf)
    segsum[t] = 0.0f;
  }
}

// ------------------------------------------------- fused QKV projection GEMM
// q = X @ Wq^T + bq (and k, v). TDM stages the 16x128 X tile into LDS once per
// block; each wave owns one 16x16 output tile and reuses the LDS A fragment
// across three v_wmma_f32_16x16x4_f32 accumulations per k-step.
__global__ void __launch_bounds__(256) gat_qkv(
    const float* __restrict__ X,
    const float* __restrict__ Wq, const float* __restrict__ bq,
    const float* __restrict__ Wk, const float* __restrict__ bk,
    const float* __restrict__ Wv, const float* __restrict__ bv,
    float* __restrict__ q, float* __restrict__ k, float* __restrict__ v) {
  __shared__ float xtile[16 * NFEAT];
  LDS_KEEPALIVE(xtile);

  if (threadIdx.x < 32) {
    tdm_load_tile16x128(X + (size_t)blockIdx.x * 16 * NFEAT);
  }
  __syncthreads();

  const int wave  = threadIdx.x >> 5;   // col tile 0..7 (features)
  const int lane  = threadIdx.x & 31;
  const int m     = lane & 15;
  const int khalf = lane >> 4;          // lanes 16-31 carry K+2
  const size_t fcol = (size_t)wave * 16 + m;            // B: output feature
  const float* xl = xtile + m * NFEAT + 2 * khalf;      // A from LDS
  const float* wq = Wq + fcol * NFEAT + 2 * khalf;
  const float* wk = Wk + fcol * NFEAT + 2 * khalf;
  const float* wv = Wv + fcol * NFEAT + 2 * khalf;

  v8f cq = {}; v8f ck = {}; v8f cv = {};
#pragma unroll 4
  for (int kb = 0; kb < NFEAT; kb += 4) {
    v2f a  = *(const v2f*)(xl + kb);
    v2f b0 = *(const v2f*)(wq + kb);
    v2f b1 = *(const v2f*)(wk + kb);
    v2f b2 = *(const v2f*)(wv + kb);
    cq = __builtin_amdgcn_wmma_f32_16x16x4_f32(false, a, false, b0, (short)0, cq, false, false);
    ck = __builtin_amdgcn_wmma_f32_16x16x4_f32(false, a, false, b1, (short)0, ck, false, false);
    cv = __builtin_amdgcn_wmma_f32_16x16x4_f32(false, a, false, b2, (short)0, cv, false, false);
  }
  const float zq = bq[wave * 16 + m];
  const float zk = bk[wave * 16 + m];
  const float zv = bv[wave * 16 + m];
#pragma unroll
  for (int r = 0; r < 8; ++r) {
    size_t orow = (size_t)blockIdx.x * 16 + r + 8 * khalf;
    size_t idx  = orow * NFEAT + (size_t)wave * 16 + m;
    q[idx] = cq[r] + zq;
    k[idx] = ck[r] + zk;
    v[idx] = cv[r] + zv;
  }
}

// ----------------------------------------------- per-edge scores + seg max
__global__ void gat_scores(const float* __restrict__ q, const float* __restrict__ k,
                           const int* __restrict__ ei, float* __restrict__ sc,
                           unsigned* __restrict__ segmax) {
  int t = blockIdx.x * blockDim.x + threadIdx.x;
  if (t >= NEDGES * NHEAD) return;
  int e = t >> 3, h = t & 7;
  int src = ei[e];
  int dst = ei[NEDGES + e];
  const float4* qp = (const float4*)(q + (size_t)src * NFEAT + h * 16);
  const float4* kp = (const float4*)(k + (size_t)dst * NFEAT + h * 16);
  float s = 0.0f;
#pragma unroll
  for (int i = 0; i < 4; ++i) {
    float4 a = qp[i], b = kp[i];
    s += a.x * b.x + a.y * b.y + a.z * b.z + a.w * b.w;
  }
  s *= 0.25f;  // 1/sqrt(D), D=16
  sc[t] = s;
  atomicMax(segmax + (size_t)src * NHEAD + h, f2ord(s));
}

// ----------------------------------------------- exp(score - max) + seg sum
__global__ void gat_exp(float* __restrict__ sc, const unsigned* __restrict__ segmax,
                        float* __restrict__ segsum, const int* __restrict__ ei) {
  int t = blockIdx.x * blockDim.x + threadIdx.x;
  if (t >= NEDGES * NHEAD) return;
  int e = t >> 3, h = t & 7;
  int src = ei[e];
  float mx = ord2f(segmax[(size_t)src * NHEAD + h]);
  float ex = __expf(sc[t] - mx);
  sc[t] = ex;
  atomicAdd(segsum + (size_t)src * NHEAD + h, ex);
}

// ----------------------------------------------- weighted-V scatter to dest
// one wave per edge; lane handles 4 contiguous features (head = lane/4)
__global__ void __launch_bounds__(256) gat_scatter(
    const float* __restrict__ v, const float* __restrict__ sc,
    const float* __restrict__ segsum, const int* __restrict__ ei,
    float* __restrict__ agg) {
  int wave = threadIdx.x >> 5;
  int lane = threadIdx.x & 31;
  int e = blockIdx.x * 8 + wave;
  if (e >= NEDGES) return;
  int src = ei[e];
  int dst = ei[NEDGES + e];
  int h = lane >> 2;
  float w = sc[(size_t)e * NHEAD + h] / segsum[(size_t)src * NHEAD + h];
  float4 val = *(const float4*)(v + (size_t)src * NFEAT + lane * 4);
  float* ap = agg + (size_t)dst * NFEAT + lane * 4;
  atomicAdd(ap + 0, val.x * w);
  atomicAdd(ap + 1, val.y * w);
  atomicAdd(ap + 2, val.z * w);
  atomicAdd(ap + 3, val.w * w);
}

// ------------------------------------------------- output projection GEMM
__global__ void __launch_bounds__(256) gat_out(
    const float* __restrict__ A, const float* __restrict__ Wo,
    const float* __restrict__ bo, float* __restrict__ out) {
  __shared__ float atile[16 * NFEAT];
  LDS_KEEPALIVE(atile);

  if (threadIdx.x < 32) {
    tdm_load_tile16x128(A + (size_t)blockIdx.x * 16 * NFEAT);
  }
  __syncthreads();

  const int wave  = threadIdx.x >> 5;
  const int lane  = threadIdx.x & 31;
  const int m     = lane & 15;
  const int khalf = lane >> 4;
  const size_t fcol = (size_t)wave * 16 + m;
  const float* al = atile + m * NFEAT + 2 * khalf;
  const float* wr = Wo + fcol * NFEAT + 2 * khalf;

  v8f c = {};
#pragma unroll 4
  for (int kb = 0; kb < NFEAT; kb += 4) {
    v2f a = *(const v2f*)(al + kb);
    v2f b = *(const v2f*)(wr + kb);
    c = __builtin_amdgcn_wmma_f32_16x16x4_f32(false, a, false, b, (short)0, c, false, false);
  }
  const float z = bo[wave * 16 + m];
#pragma unroll
  for (int r = 0; r < 8; ++r) {
    size_t orow = (size_t)blockIdx.x * 16 + r + 8 * khalf;
    out[orow * NFEAT + (size_t)wave * 16 + m] = c[r] + z;
  }
}

extern "C" void kernel_launch(void* const* d_in, const int* in_sizes, int n_in,
                              void* d_out, int out_size, void* d_ws, size_t ws_size,
                              hipStream_t stream) {
  const float* X  = (const float*)d_in[0];
  const int*   ei = (const int*)d_in[1];
  const float* Wq = (const float*)d_in[2];
  const float* bq = (const float*)d_in[3];
  const float* Wk = (const float*)d_in[4];
  const float* bk = (const float*)d_in[5];
  const float* Wv = (const float*)d_in[6];
  const float* bv = (const float*)d_in[7];
  const float* Wo = (const float*)d_in[8];
  const float* bo = (const float*)d_in[9];
  float* out = (float*)d_out;

  const size_t NF = (size_t)NNODES * NFEAT;
  float* q   = (float*)d_ws;
  float* k   = q + NF;
  float* v   = k + NF;
  float* agg = v + NF;
  float* sc  = agg + NF;                                   // E*H floats
  unsigned* segmax = (unsigned*)(sc + (size_t)NEDGES * NHEAD);
  float*    segsum = (float*)(segmax + (size_t)NNODES * NHEAD);

  gat_init<<<(NNODES * NFEAT + 255) / 256, 256, 0, stream>>>(agg, segmax, segsum);
  gat_qkv<<<NNODES / 16, 256, 0, stream>>>(X, Wq, bq, Wk, bk, Wv, bv, q, k, v);
  gat_scores<<<(NEDGES * NHEAD + 255) / 256, 256, 0, stream>>>(q, k, ei, sc, segmax);
  gat_exp<<<(NEDGES * NHEAD + 255) / 256, 256, 0, stream>>>(sc, segmax, segsum, ei);
  gat_scatter<<<NEDGES / 8, 256, 0, stream>>>(v, sc, segsum, ei, agg);
  gat_out<<<NNODES / 16, 256, 0, stream>>>(agg, Wo, bo, out);
}